// BiLSTM_CRF_76570676953698
// MI455X (gfx1250) — compile-verified
//
#include <hip/hip_runtime.h>
#include <math.h>

// ---------------------------------------------------------------------------
// BiLSTM-CRF forward for MI455X (gfx1250), fp32 WMMA (V_WMMA_F32_16X16X4_F32)
// 2x2 register-blocked tiles: 4 WMMAs per 4 loads (1 load/WMMA).
// ---------------------------------------------------------------------------
typedef float v2f __attribute__((ext_vector_type(2)));
typedef float v8f __attribute__((ext_vector_type(8)));

#define S_LEN 256
#define BATCH 32
#define EMB   512
#define HID   512
#define G4    2048   // 4*HID gate columns
#define NTAG  30
#define TAG_START 28
#define TAG_STOP  29

__device__ __forceinline__ v8f wmma4(v2f a, v2f b, v8f c) {
  // 8 args: (neg_a, A, neg_b, B, c_mod, C, reuse_a, reuse_b)
  return __builtin_amdgcn_wmma_f32_16x16x4_f32(false, a, false, b, (short)0, c,
                                               false, false);
}

__device__ __forceinline__ float sigf(float x) { return 1.0f / (1.0f + expf(-x)); }

// --------------------------- init: h state + counters ----------------------
__global__ void k_init(const float* __restrict__ h0, float* __restrict__ h_g,
                       int* __restrict__ cnt) {
  int tid = blockIdx.x * blockDim.x + threadIdx.x;
  if (tid < 2 * S_LEN) cnt[tid] = 0;
  for (int i = tid; i < 2 * BATCH * HID; i += gridDim.x * blockDim.x) {
    int d = i / (BATCH * HID);
    int r = i % (BATCH * HID);
    h_g[(size_t)(d * 2 + 0) * (BATCH * HID) + r] = h0[i];  // parity 0
  }
}

// ------------------- phase 1: embed-gather + input projection --------------
// P[d][s*B+b][g] = sum_e emb[sentence[s,b]][e] * w_ih_d[g][e] + b_ih_d[g] + b_hh_d[g]
// Each wave: one 32x32 macro-tile (2x2 blocking of 16x16 WMMA tiles), K=512.
__global__ __launch_bounds__(256) void k_proj(
    const int* __restrict__ sentence, const float* __restrict__ emb,
    const float* __restrict__ w_ih_f, const float* __restrict__ b_ih_f,
    const float* __restrict__ b_hh_f, const float* __restrict__ w_ih_b,
    const float* __restrict__ b_ih_b, const float* __restrict__ b_hh_b,
    float* __restrict__ P) {
  int wave = threadIdx.x >> 5;
  int lane = threadIdx.x & 31;
  int lo   = lane & 15;
  int half = lane >> 4;

  int tile = blockIdx.x * 8 + wave;   // 0 .. 32767 macro-tiles
  int d    = tile >> 14;              // 16384 macro-tiles / direction
  int r    = tile & 0x3FFF;
  int mt   = r >> 6;                  // 0..255  (M = 8192 rows / 32)
  int nt   = r & 63;                  // 0..63   (N = 2048 cols / 32)

  int r0 = mt * 32 + lo;              // A rows (same for both halves)
  int r1 = r0 + 16;
  const float* arow0 = emb + (size_t)sentence[r0] * EMB;
  const float* arow1 = emb + (size_t)sentence[r1] * EMB;

  int g0 = nt * 32 + lo;              // B cols
  int g1 = g0 + 16;
  const float* wih   = d ? w_ih_b : w_ih_f;
  const float* brow0 = wih + (size_t)g0 * EMB;
  const float* brow1 = wih + (size_t)g1 * EMB;

  v8f acc00 = {0.f, 0.f, 0.f, 0.f, 0.f, 0.f, 0.f, 0.f};
  v8f acc01 = acc00, acc10 = acc00, acc11 = acc00;
#pragma unroll 4
  for (int k0 = 0; k0 < EMB; k0 += 4) {
    int ko = k0 + 2 * half;
    v2f a0 = *(const v2f*)(arow0 + ko);
    v2f a1 = *(const v2f*)(arow1 + ko);
    v2f b0 = *(const v2f*)(brow0 + ko);
    v2f b1 = *(const v2f*)(brow1 + ko);
    acc00 = wmma4(a0, b0, acc00);
    acc01 = wmma4(a0, b1, acc01);
    acc10 = wmma4(a1, b0, acc10);
    acc11 = wmma4(a1, b1, acc11);
  }

  float bias0 = d ? (b_ih_b[g0] + b_hh_b[g0]) : (b_ih_f[g0] + b_hh_f[g0]);
  float bias1 = d ? (b_ih_b[g1] + b_hh_b[g1]) : (b_ih_f[g1] + b_hh_f[g1]);
  float* Pd = P + (size_t)d * (S_LEN * BATCH) * G4;
#pragma unroll
  for (int v = 0; v < 8; ++v) {
    size_t rowA = (size_t)(mt * 32 + v + 8 * half);
    Pd[rowA * G4 + g0]        = acc00[v] + bias0;
    Pd[rowA * G4 + g1]        = acc01[v] + bias1;
    Pd[(rowA + 16) * G4 + g0] = acc10[v] + bias0;
    Pd[(rowA + 16) * G4 + g1] = acc11[v] + bias1;
  }
}

// ------------------- phase 2: persistent LSTM recurrence -------------------
// 16 blocks: (direction d, hidden slice jb of 64 cols). Per step each block
// does g = h_prev @ W_hh_sliceT via WMMA (each wave: full batch M=32 x its
// 32 gate cols, 2x2 blocked), applies LSTM pointwise math for its slice,
// publishes h to a parity-double-buffered global buffer, then joins a
// device-scope spin barrier with the other 7 blocks of its direction.
__global__ __launch_bounds__(256) void k_lstm(
    const float* __restrict__ P, const float* __restrict__ w_hh_f,
    const float* __restrict__ w_hh_b, const float* __restrict__ c0,
    float* __restrict__ h_g, int* cnt, float* __restrict__ lstm_out) {
  __shared__ float gsm[BATCH * 256];  // gate tiles for this slice (32 x 256)
  __shared__ float cs[BATCH * 64];    // cell state slice

  int tid = threadIdx.x;
  int blk = blockIdx.x;
  int d   = blk >> 3;
  int jb  = blk & 7;
  int J0  = jb * 64;
  const float* whh = d ? w_hh_b : w_hh_f;
  const float* Pd  = P + (size_t)d * (S_LEN * BATCH) * G4;

  for (int idx = tid; idx < BATCH * 64; idx += 256) {
    int b = idx >> 6, j = idx & 63;
    cs[idx] = c0[((size_t)d * BATCH + b) * HID + J0 + j];
  }
  int wave = tid >> 5, lane = tid & 31, lo = lane & 15, half = lane >> 4;

  // Loop-invariant B (w_hh) row pointers: wave owns local gate cols
  // [wave*32, wave*32+32); n0 = wave*32+lo, n1 = n0+16.
  int n0 = wave * 32 + lo;
  int n1 = n0 + 16;
  int gate0 = n0 >> 6, j0 = n0 & 63;
  int gate1 = n1 >> 6, j1 = n1 & 63;
  const float* brow0 = whh + (size_t)(gate0 * HID + J0 + j0) * HID;
  const float* brow1 = whh + (size_t)(gate1 * HID + J0 + j1) * HID;

  const float* hbuf0 = h_g + (size_t)(d * 2 + 0) * (BATCH * HID);
  const float* hbuf1 = h_g + (size_t)(d * 2 + 1) * (BATCH * HID);
  __syncthreads();

  for (int t = 0; t < S_LEN; ++t) {
    int par = t & 1;
    const float* hin = par ? hbuf1 : hbuf0;
    const float* arow0 = hin + (size_t)lo * HID;         // batch rows 0..15
    const float* arow1 = hin + (size_t)(lo + 16) * HID;  // batch rows 16..31

    v8f acc00 = {0.f, 0.f, 0.f, 0.f, 0.f, 0.f, 0.f, 0.f};
    v8f acc01 = acc00, acc10 = acc00, acc11 = acc00;
#pragma unroll 4
    for (int k0 = 0; k0 < HID; k0 += 4) {
      int ko = k0 + 2 * half;
      v2f a0 = *(const v2f*)(arow0 + ko);
      v2f a1 = *(const v2f*)(arow1 + ko);
      v2f b0 = *(const v2f*)(brow0 + ko);
      v2f b1 = *(const v2f*)(brow1 + ko);
      acc00 = wmma4(a0, b0, acc00);
      acc01 = wmma4(a0, b1, acc01);
      acc10 = wmma4(a1, b0, acc10);
      acc11 = wmma4(a1, b1, acc11);
    }
#pragma unroll
    for (int v = 0; v < 8; ++v) {
      int rw = v + 8 * half;
      gsm[rw * 256 + n0]        = acc00[v];
      gsm[rw * 256 + n1]        = acc01[v];
      gsm[(rw + 16) * 256 + n0] = acc10[v];
      gsm[(rw + 16) * 256 + n1] = acc11[v];
    }
    __syncthreads();

    // --- pointwise LSTM update for this hidden slice -----------------------
    int s_eff = d ? (S_LEN - 1 - t) : t;
    float* hout = h_g + (size_t)(d * 2 + (par ^ 1)) * (BATCH * HID);
    for (int idx = tid; idx < BATCH * 64; idx += 256) {
      int b = idx >> 6, j = idx & 63;
      const float* Prow = Pd + ((size_t)s_eff * BATCH + b) * G4;
      float gi = gsm[b * 256 +       j] + Prow[          J0 + j];
      float gf = gsm[b * 256 +  64 + j] + Prow[    HID + J0 + j];
      float gg = gsm[b * 256 + 128 + j] + Prow[2 * HID + J0 + j];
      float go = gsm[b * 256 + 192 + j] + Prow[3 * HID + J0 + j];
      float c  = sigf(gf) * cs[idx] + sigf(gi) * tanhf(gg);
      cs[idx]  = c;
      float h  = sigf(go) * tanhf(c);
      hout[b * HID + J0 + j] = h;
      lstm_out[((size_t)s_eff * BATCH + b) * (2 * HID) + d * HID + J0 + j] = h;
    }
    __threadfence();
    __syncthreads();
    if (tid == 0) {
      __hip_atomic_fetch_add(&cnt[d * S_LEN + t], 1, __ATOMIC_ACQ_REL,
                             __HIP_MEMORY_SCOPE_AGENT);
      while (__hip_atomic_load(&cnt[d * S_LEN + t], __ATOMIC_ACQUIRE,
                               __HIP_MEMORY_SCOPE_AGENT) < 8) {}
    }
    __syncthreads();
    __threadfence();  // acquire: invalidate L0/L1 so next step sees peers' h
  }
}

// ------------------- phase 3: output projection (feats) --------------------
__global__ void k_feats(const float* __restrict__ lstm_out,
                        const float* __restrict__ w_out,
                        const float* __restrict__ b_out,
                        float* __restrict__ feats) {
  int sb = blockIdx.x;   // 0..8191  (s*B+b)
  int t  = threadIdx.x;  // 32 threads, 30 active
  if (t >= NTAG) return;
  const float* row = lstm_out + (size_t)sb * (2 * HID);
  const float* w   = w_out + (size_t)t * (2 * HID);
  float acc = b_out[t];
  for (int h = 0; h < 2 * HID; ++h) acc += row[h] * w[h];
  int s = sb / BATCH, b = sb % BATCH;
  feats[((size_t)b * S_LEN + s) * NTAG + t] = acc;  // (B,S,T) layout
}

// ------------------- phase 4: CRF forward scan + gold + reduce -------------
__global__ __launch_bounds__(1024) void k_crf(const float* __restrict__ feats,
                                              const float* __restrict__ trans,
                                              const int* __restrict__ tags,
                                              float* __restrict__ out) {
  __shared__ float fv[BATCH * NTAG];
  __shared__ float fv2[BATCH * NTAG];
  __shared__ float tr[NTAG * NTAG];
  __shared__ float diff[BATCH];
  int tid = threadIdx.x;
  for (int i = tid; i < NTAG * NTAG; i += 1024) tr[i] = trans[i];
  int b = tid / NTAG, t = tid % NTAG;
  bool valid = tid < BATCH * NTAG;
  if (valid) fv[tid] = (t == TAG_START) ? 0.0f : -10000.0f;
  __syncthreads();

  for (int s = 0; s < S_LEN; ++s) {
    if (valid) {
      float m = -3.4e38f;
      for (int t2 = 0; t2 < NTAG; ++t2)
        m = fmaxf(m, fv[b * NTAG + t2] + tr[t * NTAG + t2]);
      float ssum = 0.f;
      for (int t2 = 0; t2 < NTAG; ++t2)
        ssum += expf(fv[b * NTAG + t2] + tr[t * NTAG + t2] - m);
      fv2[tid] = m + logf(ssum) + feats[((size_t)b * S_LEN + s) * NTAG + t];
    }
    __syncthreads();
    if (valid) fv[tid] = fv2[tid];
    __syncthreads();
  }

  if (tid < BATCH) {
    int bb = tid;
    float m = -3.4e38f;
    for (int t2 = 0; t2 < NTAG; ++t2)
      m = fmaxf(m, fv[bb * NTAG + t2] + tr[TAG_STOP * NTAG + t2]);
    float ssum = 0.f;
    for (int t2 = 0; t2 < NTAG; ++t2)
      ssum += expf(fv[bb * NTAG + t2] + tr[TAG_STOP * NTAG + t2] - m);
    float fwd = m + logf(ssum);

    int   prev = TAG_START;
    float gold = 0.f;
    for (int s = 0; s < S_LEN; ++s) {
      int nxt = tags[s * BATCH + bb];
      gold += tr[nxt * NTAG + prev] + feats[((size_t)bb * S_LEN + s) * NTAG + nxt];
      prev = nxt;
    }
    gold += tr[TAG_STOP * NTAG + prev];
    diff[bb] = fwd - gold;
  }
  __syncthreads();
  if (tid == 0) {
    float s = 0.f;
    for (int i = 0; i < BATCH; ++i) s += diff[i];
    out[0] = s;
  }
}

// ---------------------------------------------------------------------------
extern "C" void kernel_launch(void* const* d_in, const int* in_sizes, int n_in,
                              void* d_out, int out_size, void* d_ws,
                              size_t ws_size, hipStream_t stream) {
  const int*   sentence = (const int*)d_in[0];
  const int*   tags     = (const int*)d_in[1];
  const float* emb      = (const float*)d_in[2];
  const float* w_ih_f   = (const float*)d_in[3];
  const float* w_hh_f   = (const float*)d_in[4];
  const float* b_ih_f   = (const float*)d_in[5];
  const float* b_hh_f   = (const float*)d_in[6];
  const float* w_ih_b   = (const float*)d_in[7];
  const float* w_hh_b   = (const float*)d_in[8];
  const float* b_ih_b   = (const float*)d_in[9];
  const float* b_hh_b   = (const float*)d_in[10];
  const float* w_out    = (const float*)d_in[11];
  const float* b_out    = (const float*)d_in[12];
  const float* trans    = (const float*)d_in[13];
  const float* h0       = (const float*)d_in[14];
  const float* c0       = (const float*)d_in[15];
  float* out = (float*)d_out;

  // workspace carve-up (float units)
  size_t nP = (size_t)2 * S_LEN * BATCH * G4;      // 33,554,432  (128 MB)
  size_t nL = (size_t)S_LEN * BATCH * 2 * HID;     //  8,388,608  ( 32 MB)
  size_t nF = (size_t)BATCH * S_LEN * NTAG;        //    245,760
  size_t nH = (size_t)2 * 2 * BATCH * HID;         //     65,536
  size_t offP = 0;
  size_t offL = offP + nP;
  size_t offF = offL + nL;
  size_t offH = offF + nF;
  size_t offC = offH + nH;
  size_t needed = (offC + 2 * S_LEN) * sizeof(float);
  if (ws_size < needed) return;  // deterministic no-op if scratch too small

  float* ws    = (float*)d_ws;
  float* P     = ws + offP;
  float* lstm  = ws + offL;
  float* feats = ws + offF;
  float* h_g   = ws + offH;
  int*   cnt   = (int*)(ws + offC);

  k_init<<<64, 256, 0, stream>>>(h0, h_g, cnt);
  k_proj<<<4096, 256, 0, stream>>>(sentence, emb, w_ih_f, b_ih_f, b_hh_f,
                                   w_ih_b, b_ih_b, b_hh_b, P);
  k_lstm<<<16, 256, 0, stream>>>(P, w_hh_f, w_hh_b, c0, h_g, cnt, lstm);
  k_feats<<<S_LEN * BATCH, 32, 0, stream>>>(lstm, w_out, b_out, feats);
  k_crf<<<1, 1024, 0, stream>>>(feats, trans, tags, out);
}